// GINConv_47974784697090
// MI455X (gfx1250) — compile-verified
//
#include <hip/hip_runtime.h>

typedef __attribute__((ext_vector_type(2))) float v2f;
typedef __attribute__((ext_vector_type(8))) float v8f;

#define LDSROW 68   // padded row stride (words) -> conflict-free b64 A-fragment loads

// ---------------------------------------------------------------------------
// Kernel 1: init agg to -inf sentinel
// ---------------------------------------------------------------------------
__global__ __launch_bounds__(256) void init_agg_kernel(float* agg, int total) {
  int i = blockIdx.x * 256 + threadIdx.x;
  if (i < total) ((unsigned int*)agg)[i] = 0xFF800000u;  // -inf
}

// ---------------------------------------------------------------------------
// Kernel 2: pre-swizzle W1/W2 into WMMA B-fragment order.
// B[k][n] = W[n][k] (we compute x @ W^T). Fragment (nt,kt): lane (halfk,nlane)
// loads float2 = ( W[nt*16+nlane][kt*4+2*halfk], W[nt*16+nlane][kt*4+2*halfk+1] )
// stored contiguously so each wave fragment load is one 256B coalesced read.
// ---------------------------------------------------------------------------
__global__ __launch_bounds__(256) void stage_weights_kernel(const float* __restrict__ W1,
                                                            const float* __restrict__ W2,
                                                            float* __restrict__ sW1,
                                                            float* __restrict__ sW2) {
  int tid = blockIdx.x * 256 + threadIdx.x;  // 0..4095
  int j    =  tid        & 1;
  int n    = (tid >> 1)  & 15;
  int half = (tid >> 5)  & 1;
  int kt   = (tid >> 6)  & 15;
  int nt   =  tid >> 10;
  int s = (nt * 16 + n) * 64 + kt * 4 + half * 2 + j;
  sW1[tid] = W1[s];
  sW2[tid] = W2[s];
}

// ---------------------------------------------------------------------------
// Kernel 3: edge scatter-max. One wave per edge: coalesced 2x128B read of
// h[src], hardware atomics into agg[dst] (agg fits in L2).
// Float max via monotone int mapping: no CAS loop.
// ---------------------------------------------------------------------------
__device__ __forceinline__ void atomic_max_float(float* addr, float val) {
  val = val + 0.0f;  // canonicalize -0.0 -> +0.0
  if (val >= 0.0f)
    atomicMax((int*)addr, __float_as_int(val));           // global_atomic_max_i32
  else
    atomicMin((unsigned int*)addr, __float_as_uint(val)); // global_atomic_min_u32
}

__global__ __launch_bounds__(256) void edge_max_kernel(const float* __restrict__ h,
                                                       const int* __restrict__ src,
                                                       const int* __restrict__ dst,
                                                       float* __restrict__ agg, int E) {
  int wv   = blockIdx.x * 8 + (threadIdx.x >> 5);
  int lane = threadIdx.x & 31;
  if (wv >= E) return;
  int s = src[wv];
  int d = dst[wv];
  const float* hs = h   + (size_t)s * 64;
  float*       ad = agg + (size_t)d * 64;
  float v0 = hs[lane];
  float v1 = hs[lane + 32];
  atomic_max_float(ad + lane, v0);
  atomic_max_float(ad + lane + 32, v1);
}

// ---------------------------------------------------------------------------
// Kernel 4: fused  out = relu((h + agg') @ W1^T) @ W2^T + b2
// One wave per 16-node tile; fp32 WMMA 16x16x4, 16 k-steps per 16x16 tile,
// 4 column tiles per GEMM -> 128 v_wmma per tile.
// ---------------------------------------------------------------------------
__global__ __launch_bounds__(32) void gin_mlp_kernel(const float* __restrict__ h,
                                                     const float* __restrict__ agg,
                                                     const float* __restrict__ sW1,
                                                     const float* __restrict__ sW2,
                                                     const float* __restrict__ b2,
                                                     float* __restrict__ out) {
  __shared__ float xs[16 * LDSROW];
  __shared__ float ys[16 * LDSROW];
  const int lane  = threadIdx.x;
  const int halfk = lane >> 4;   // 0: K+{0,1}, rows M=0..7 of C; 1: K+{2,3}, rows M=8..15
  const int nlane = lane & 15;
  const size_t base = (size_t)blockIdx.x * 16 * 64;

  // ---- stage x = h + (agg == -inf ? 0 : agg) into LDS (16 x 64, stride 68) ----
  const float4* h4 = (const float4*)(h + base);
  const float4* a4 = (const float4*)(agg + base);
#pragma unroll
  for (int i = 0; i < 8; ++i) {
    int idx = i * 32 + lane;          // float4 index 0..255
    float4 hv = h4[idx];
    float4 av = a4[idx];
    int m = idx >> 4;
    int c = (idx & 15) * 4;
    float4 xv;
    xv.x = hv.x + (av.x < -1e30f ? 0.0f : av.x);
    xv.y = hv.y + (av.y < -1e30f ? 0.0f : av.y);
    xv.z = hv.z + (av.z < -1e30f ? 0.0f : av.z);
    xv.w = hv.w + (av.w < -1e30f ? 0.0f : av.w);
    *(float4*)&xs[m * LDSROW + c] = xv;
  }
  __syncthreads();

  // ---- GEMM1: y = relu(x @ W1^T) ----
  const float2* w1p = (const float2*)sW1;
#pragma unroll
  for (int nt = 0; nt < 4; ++nt) {
    v8f acc = {};
#pragma unroll
    for (int kt = 0; kt < 16; ++kt) {
      float2 bf = w1p[((nt * 16 + kt) * 2 + halfk) * 16 + nlane];
      float2 af = *(const float2*)&xs[nlane * LDSROW + kt * 4 + halfk * 2];
      v2f a; a[0] = af.x; a[1] = af.y;
      v2f b; b[0] = bf.x; b[1] = bf.y;
      acc = __builtin_amdgcn_wmma_f32_16x16x4_f32(false, a, false, b,
                                                  (short)0, acc, false, false);
    }
#pragma unroll
    for (int i = 0; i < 8; ++i) {
      int m = i + 8 * halfk;
      float v = acc[i];
      ys[m * LDSROW + nt * 16 + nlane] = v > 0.0f ? v : 0.0f;
    }
  }
  __syncthreads();

  // ---- GEMM2: out = y @ W2^T + b2 ----
  const float2* w2p = (const float2*)sW2;
#pragma unroll
  for (int nt = 0; nt < 4; ++nt) {
    v8f acc = {};
#pragma unroll
    for (int kt = 0; kt < 16; ++kt) {
      float2 bf = w2p[((nt * 16 + kt) * 2 + halfk) * 16 + nlane];
      float2 af = *(const float2*)&ys[nlane * LDSROW + kt * 4 + halfk * 2];
      v2f a; a[0] = af.x; a[1] = af.y;
      v2f b; b[0] = bf.x; b[1] = bf.y;
      acc = __builtin_amdgcn_wmma_f32_16x16x4_f32(false, a, false, b,
                                                  (short)0, acc, false, false);
    }
    float bias = b2[nt * 16 + nlane];
#pragma unroll
    for (int i = 0; i < 8; ++i) {
      int m = i + 8 * halfk;
      out[base + (size_t)m * 64 + nt * 16 + nlane] = acc[i] + bias;
    }
  }
}

// ---------------------------------------------------------------------------
extern "C" void kernel_launch(void* const* d_in, const int* in_sizes, int n_in,
                              void* d_out, int out_size, void* d_ws, size_t ws_size,
                              hipStream_t stream) {
  const float* h  = (const float*)d_in[0];
  const int*   src = (const int*)d_in[1];
  const int*   dst = (const int*)d_in[2];
  const float* W1 = (const float*)d_in[3];
  const float* W2 = (const float*)d_in[4];
  const float* b2 = (const float*)d_in[5];
  const int D = 64;
  const int N = in_sizes[0] / D;
  const int E = in_sizes[1];

  // workspace layout: agg[N*D] | sW1[4096] | sW2[4096]
  float* agg = (float*)d_ws;
  float* sW1 = agg + (size_t)N * D;
  float* sW2 = sW1 + D * D;

  int total = N * D;
  init_agg_kernel<<<(total + 255) / 256, 256, 0, stream>>>(agg, total);
  stage_weights_kernel<<<16, 256, 0, stream>>>(W1, W2, sW1, sW2);
  edge_max_kernel<<<(E + 7) / 8, 256, 0, stream>>>(h, src, dst, agg, E);
  gin_mlp_kernel<<<N / 16, 32, 0, stream>>>(h, agg, sW1, sW2, b2, (float*)d_out);
}